// FFFInference_73169062855234
// MI455X (gfx1250) — compile-verified
//
#include <hip/hip_runtime.h>

typedef float v2f __attribute__((ext_vector_type(2)));
typedef float v4f __attribute__((ext_vector_type(4)));
typedef float v8f __attribute__((ext_vector_type(8)));
typedef int   v4i __attribute__((ext_vector_type(4)));

#define TREE_DEPTH 4
#define N_NODES    31
#define D_IN       1024
#define D_OUT      1024
#define ROWS_PER_WG 128
#define KCHUNK      128
#define NCHUNK      (D_IN / KCHUNK)   // 8

// LDS strides padded so row offset is 16B-aligned (b128 transfers) and rows
// land on distinct banks (stride mod 64 banks == +4) for the 16-lane reads.
#define LDSX_STRIDE 132   // x chunk rows: 132*4 = 528 B
#define W_STRIDE    1028  // W1/W2 rows:  1028*4 = 4112 B
#define C_STRIDE    34    // coeff rows

#define SW1_OFF 0
#define SX0_OFF (32 * W_STRIDE)                       // 32896 floats
#define SX_BUF  (ROWS_PER_WG * LDSX_STRIDE)           // 16896 floats / buffer
#define SX1_OFF (SX0_OFF + SX_BUF)                    // 49792
#define SC_OFF  (SX1_OFF + SX_BUF)                    // 66688
#define LDS_FLOATS (SC_OFF + ROWS_PER_WG * C_STRIDE)  // 71040 floats
#define LDS_BYTES  (LDS_FLOATS * 4)                   // 284160 B <= 320KB

#if defined(__has_builtin)
#if __has_builtin(__builtin_amdgcn_global_load_async_to_lds_b128)
#define ASYNC_STAGE 1
#endif
#endif
#ifndef ASYNC_STAGE
#define ASYNC_STAGE 0
#endif

__device__ __forceinline__ float gelu_tanh(float v) {
    // jax.nn.gelu default (approximate=True)
    float v3 = v * v * v;
    return 0.5f * v * (1.0f + tanhf(0.7978845608028654f * (v + 0.044715f * v3)));
}

#if ASYNC_STAGE
// builtin wants int4-vector pointers: (global src, lds dst, imm offset, imm cpol)
typedef __attribute__((address_space(1))) v4i* gv4i_p;
typedef __attribute__((address_space(3))) v4i* lv4i_p;

__device__ __forceinline__ void async_ld_b128(const float* g, float* l) {
    __builtin_amdgcn_global_load_async_to_lds_b128(
        (gv4i_p)(g), (lv4i_p)(l), 0, 0);
}
__device__ __forceinline__ void wait_async_le16() {
#if __has_builtin(__builtin_amdgcn_s_wait_asynccnt)
    __builtin_amdgcn_s_wait_asynccnt(16);
#else
    asm volatile("s_wait_asynccnt 16" ::: "memory");
#endif
}
__device__ __forceinline__ void wait_async_le0() {
#if __has_builtin(__builtin_amdgcn_s_wait_asynccnt)
    __builtin_amdgcn_s_wait_asynccnt(0);
#else
    asm volatile("s_wait_asynccnt 0" ::: "memory");
#endif
}
#endif

__global__ __launch_bounds__(256)
void fff_fused_kernel(const float* __restrict__ x,
                      const float* __restrict__ W1,
                      const float* __restrict__ W2,
                      float* __restrict__ out) {
    extern __shared__ float smem[];
    float* sW1 = smem + SW1_OFF;   // W1 padded to 32 rows, persistent
    float* sX0 = smem + SX0_OFF;   // x chunk double-buffer (reused as W2 later)
    float* sX1 = smem + SX1_OFF;
    float* sC  = smem + SC_OFF;    // vals then coefficients [128][34]
    float* sW2 = sX0;              // phase 2: W2 spans both x buffers

    const int tid     = threadIdx.x;
    const int lane    = tid & 31;
    const int wave    = tid >> 5;
    const int rowBase = blockIdx.x * ROWS_PER_WG;

    // wave32 WMMA f32 fragment decomposition (16x4 A / 4x16 B / 16x16 C)
    const int lr   = lane & 15;   // M (A) / N (B,C/D)
    const int hi   = lane >> 4;   // half-wave select
    const int koff = hi << 1;     // K base within fragment: 0 or 2

    // ---------------- load W1 (31 x 1024) into LDS, zero pad row 31 ---------
    #pragma unroll 4
    for (int j = 0; j < 31; ++j) {
        int linear = j * 256 + tid;          // float4 index
        int r  = linear >> 8;                // 256 float4 per row
        int c4 = linear & 255;
        v4f v = *(const v4f*)(W1 + (size_t)r * D_IN + c4 * 4);
        *(v4f*)(sW1 + r * W_STRIDE + c4 * 4) = v;
    }
    {
        v4f z = {0.f, 0.f, 0.f, 0.f};
        *(v4f*)(sW1 + 31 * W_STRIDE + tid * 4) = z;
    }

#if ASYNC_STAGE
    // prologue: issue async global->LDS for chunk 0 (16 x b128 per thread)
    {
        float* dst = sX0;
        #pragma unroll
        for (int j = 0; j < 16; ++j) {
            int linear = j * 256 + tid;
            int r  = linear >> 5;            // 32 float4 per row
            int c4 = linear & 31;
            const float* g = x + (size_t)(rowBase + r) * D_IN + c4 * 4;
            float* l = dst + r * LDSX_STRIDE + c4 * 4;
            async_ld_b128(g, l);
        }
    }
#endif

    // ---------------- phase 1: vals[128][32] = x_tile @ W1^T ----------------
    v8f acc0 = {};   // node columns 0..15
    v8f acc1 = {};   // node columns 16..31
    for (int kc = 0; kc < NCHUNK; ++kc) {
        float* cur = (kc & 1) ? sX1 : sX0;
#if ASYNC_STAGE
        if (kc + 1 < NCHUNK) {
            float* nxt = (kc & 1) ? sX0 : sX1;
            #pragma unroll
            for (int j = 0; j < 16; ++j) {
                int linear = j * 256 + tid;
                int r  = linear >> 5;
                int c4 = linear & 31;
                const float* g = x + (size_t)(rowBase + r) * D_IN
                                   + (kc + 1) * KCHUNK + c4 * 4;
                float* l = nxt + r * LDSX_STRIDE + c4 * 4;
                async_ld_b128(g, l);
            }
            wait_async_le16();   // in-order: chunk kc's 16 loads complete
        } else {
            wait_async_le0();
        }
#else
        // synchronous staging fallback — coalesced float4
        #pragma unroll 4
        for (int j = 0; j < 16; ++j) {
            int linear = j * 256 + tid;
            int r  = linear >> 5;
            int c4 = linear & 31;
            v4f v = *(const v4f*)(x + (size_t)(rowBase + r) * D_IN
                                    + kc * KCHUNK + c4 * 4);
            *(v4f*)(cur + r * LDSX_STRIDE + c4 * 4) = v;
        }
#endif
        __syncthreads();   // chunk kc visible to all waves (W1 too on 1st pass)

        const float* xrow = cur + (wave * 16 + lr) * LDSX_STRIDE + koff;
        const float* w1n0 = sW1 + lr        * W_STRIDE + kc * KCHUNK + koff;
        const float* w1n1 = sW1 + (16 + lr) * W_STRIDE + kc * KCHUNK + koff;
        #pragma unroll 8
        for (int k = 0; k < KCHUNK; k += 4) {
            v2f a  = *(const v2f*)(xrow + k);   // A[m][k..k+1] per layout
            v2f b0 = *(const v2f*)(w1n0 + k);   // B[k][n] = W1[n][k]
            v2f b1 = *(const v2f*)(w1n1 + k);
            acc0 = __builtin_amdgcn_wmma_f32_16x16x4_f32(
                       false, a, false, b0, (short)0, acc0, false, false);
            acc1 = __builtin_amdgcn_wmma_f32_16x16x4_f32(
                       false, a, false, b1, (short)0, acc1, false, false);
        }
        __syncthreads();   // all waves done reading cur before it is re-issued
    }

    // store vals to sC: C/D layout — VGPR j, lane -> (M = j + hi*8, N = lr)
    #pragma unroll
    for (int j = 0; j < 8; ++j) {
        int r = wave * 16 + j + (hi << 3);
        sC[r * C_STRIDE + lr]      = acc0[j];
        sC[r * C_STRIDE + 16 + lr] = acc1[j];
    }

    // load W2 (31 x 1024) into the x-buffer region, zero pad row 31
    #pragma unroll 4
    for (int j = 0; j < 31; ++j) {
        int linear = j * 256 + tid;
        int r  = linear >> 8;
        int c4 = linear & 255;
        v4f v = *(const v4f*)(W2 + (size_t)r * D_OUT + c4 * 4);
        *(v4f*)(sW2 + r * W_STRIDE + c4 * 4) = v;
    }
    {
        v4f z = {0.f, 0.f, 0.f, 0.f};
        *(v4f*)(sW2 + 31 * W_STRIDE + tid * 4) = z;
    }
    __syncthreads();   // vals + W2 visible to everyone

    // ---------------- routing: one thread per batch row ---------------------
    if (tid < ROWS_PER_WG) {
        float* crow = sC + tid * C_STRIDE;
        int   pn[TREE_DEPTH + 1];
        float pv[TREE_DEPTH + 1];
        int node = 0;
        #pragma unroll
        for (int s = 0; s <= TREE_DEPTH; ++s) {
            float v = crow[node];
            pn[s] = node;
            pv[s] = gelu_tanh(v);
            node = 2 * node + 1 + (v > 0.0f ? 1 : 0);
        }
        #pragma unroll
        for (int n = 0; n < 32; ++n) crow[n] = 0.0f;
        #pragma unroll
        for (int s = 0; s <= TREE_DEPTH; ++s) crow[pn[s]] = pv[s];
    }
    __syncthreads();

    // ---------------- phase 2: out_tile = c[128][32] @ W2[32][1024] ---------
    v2f aR[8];   // hoist A fragments (coefficient row) — reused for all 64 tiles
    {
        const float* crow = sC + (wave * 16 + lr) * C_STRIDE + koff;
        #pragma unroll
        for (int i = 0; i < 8; ++i) aR[i] = *(const v2f*)(crow + i * 4);
    }
    for (int nt = 0; nt < D_OUT / 16; ++nt) {
        const int col = nt * 16 + lr;
        v8f acc = {};
        #pragma unroll
        for (int i = 0; i < 8; ++i) {
            int kk = i * 4 + koff;
            v2f b;
            b.x = sW2[kk * W_STRIDE + col];        // B[k][j] = W2[k][j]
            b.y = sW2[(kk + 1) * W_STRIDE + col];
            acc = __builtin_amdgcn_wmma_f32_16x16x4_f32(
                      false, aR[i], false, b, (short)0, acc, false, false);
        }
        #pragma unroll
        for (int j = 0; j < 8; ++j) {
            int r = rowBase + wave * 16 + j + (hi << 3);
            out[(size_t)r * D_OUT + col] = acc[j];
        }
    }
}

extern "C" void kernel_launch(void* const* d_in, const int* in_sizes, int n_in,
                              void* d_out, int out_size, void* d_ws, size_t ws_size,
                              hipStream_t stream) {
    (void)n_in; (void)out_size; (void)d_ws; (void)ws_size;
    const float* x  = (const float*)d_in[0];
    const float* W1 = (const float*)d_in[1];
    const float* W2 = (const float*)d_in[2];
    float* out = (float*)d_out;

    int batch = in_sizes[0] / D_IN;            // 65536
    dim3 grid(batch / ROWS_PER_WG);            // 512 workgroups
    dim3 block(256);                           // 8 wave32 waves
    hipLaunchKernelGGL(fff_fused_kernel, grid, block, LDS_BYTES, stream,
                       x, W1, W2, out);
}